// GraphSAGE_55490977464722
// MI455X (gfx1250) — compile-verified
//
#include <hip/hip_runtime.h>

// GraphSAGE 2-layer inference for MI455X (gfx1250).
// D_IN = D_HID = D_OUT = 64, K-dim of each GEMM = 128 (self || neighbor-agg).
#define D   64
#define DK  128

typedef float v2f __attribute__((ext_vector_type(2)));
typedef float v8f __attribute__((ext_vector_type(8)));

// ---------------------------------------------------------------- zero fill
__global__ void zero_f32(float* __restrict__ p, int n) {
    int i = blockIdx.x * blockDim.x + threadIdx.x;
    if (i < n) p[i] = 0.0f;
}

// ------------------------------------------------------- edge scatter (sum)
// 16 threads per edge, float4 (16B) per thread. Gather is one global_load_b128;
// scatter is 4 native global_atomic_add_f32 (no vector f32 atomic exists).
// Feature tables (25.6MB) are resident in the 192MB L2, so the atomics RMW in L2.
__global__ void sage_scatter(const float* __restrict__ feat,
                             const int*   __restrict__ row,
                             const int*   __restrict__ col,
                             float*       __restrict__ agg,
                             float*       __restrict__ deg,   // may be null
                             int nEdges) {
    int tid = blockIdx.x * blockDim.x + threadIdx.x;
    int e  = tid >> 4;
    int c4 = (tid & 15) << 2;
    if (e >= nEdges) return;
    int dst = row[e];
    int src = col[e];
    float4 v = *(const float4*)(feat + (size_t)src * D + c4);
    float* a = agg + (size_t)dst * D + c4;
    unsafeAtomicAdd(a + 0, v.x);
    unsafeAtomicAdd(a + 1, v.y);
    unsafeAtomicAdd(a + 2, v.z);
    unsafeAtomicAdd(a + 3, v.w);
    if (deg != nullptr && (tid & 15) == 0) unsafeAtomicAdd(&deg[dst], 1.0f);
}

// ---------------------------------------------------- WMMA fp32 GEMM layer
// out[m, 0:64] = relu?( [featSelf[m] || featAgg[m]/deg[m]] (1x128) @ W (128x64) + b )
// One wave32 computes a 16(M) x 64(N) tile with V_WMMA_F32_16X16X4_F32:
//   A 16x4 frag : lane<16 -> row M=lane,   K = k+0,k+1  (v2f, global_load_b64)
//                 lane>=16-> row M=lane-16,K = k+2,k+3
//   B 4x16 frag : lane<16 -> col N=lane,   K = k+0,k+1  (mirrored layout)
//   C/D 16x16   : 8 VGPRs, lane half selects rows M / M+8.
// W is staged in LDS *K-pair interleaved*: sW2[(k/2)*64+n] = {W[k][n],W[k+1][n]},
// so each B frag is a single aligned ds_load_b64 with no repack moves.
// Degree normalization of the aggregate is folded into the A-frag load
// (a * 1/max(deg,1)), eliminating a separate 51MB normalize pass.
__global__ void __launch_bounds__(256)
sage_gemm_wmma(const float* __restrict__ featSelf,  // [N,64] K = 0..63
               const float* __restrict__ featAgg,   // [N,64] raw sums, K = 64..127
               const float* __restrict__ deg,       // [N]
               const float* __restrict__ W,         // [128,64] row-major
               const float* __restrict__ bias,      // [64]
               float* __restrict__ out,             // [N,64]
               int nTiles, int relu) {
    __shared__ __align__(16) float2 sW2[(DK / 2) * D];   // 32 KB, K-pair interleaved

    // Cooperative interleaved W load: i -> pair p = i/64, col n = i%64.
    for (int i = threadIdx.x; i < (DK / 2) * D; i += blockDim.x) {
        int p = i >> 6;
        int n = i & 63;
        sW2[i] = make_float2(W[(2 * p) * D + n], W[(2 * p + 1) * D + n]);
    }
    __syncthreads();

    const int wave = threadIdx.x >> 5;
    const int lane = threadIdx.x & 31;
    const int tile = blockIdx.x * (blockDim.x >> 5) + wave;
    if (tile >= nTiles) return;                      // wave-uniform: EXEC stays all-1s

    const int lanelo = lane & 15;
    const int hi     = lane >> 4;                    // 0 | 1
    const int koff   = hi << 1;                      // 0 | 2 (K sub-offset of this lane)
    const int mrow   = tile * 16 + lanelo;           // A-matrix row fed by this lane

    const float* arowS = featSelf + (size_t)mrow * D;
    const float* arowA = featAgg  + (size_t)mrow * D;
    const float  rdeg  = 1.0f / fmaxf(deg[mrow], 1.0f);

    const v2f* sB = (const v2f*)sW2;

    v8f acc0 = {}, acc1 = {}, acc2 = {}, acc3 = {};

    // K = 0..63 : self features
#pragma unroll 4
    for (int k = 0; k < D; k += 4) {
        v2f a = *(const v2f*)(arowS + k + koff);
        const v2f* bk = sB + ((k >> 1) + hi) * D + lanelo;
        acc0 = __builtin_amdgcn_wmma_f32_16x16x4_f32(false, a, false, bk[ 0], (short)0, acc0, false, false);
        acc1 = __builtin_amdgcn_wmma_f32_16x16x4_f32(false, a, false, bk[16], (short)0, acc1, false, false);
        acc2 = __builtin_amdgcn_wmma_f32_16x16x4_f32(false, a, false, bk[32], (short)0, acc2, false, false);
        acc3 = __builtin_amdgcn_wmma_f32_16x16x4_f32(false, a, false, bk[48], (short)0, acc3, false, false);
    }
    // K = 64..127 : neighbor aggregate, degree-normalized on the fly
#pragma unroll 4
    for (int k = 0; k < D; k += 4) {
        v2f a = *(const v2f*)(arowA + k + koff);
        a.x *= rdeg;
        a.y *= rdeg;
        const v2f* bk = sB + (((D + k) >> 1) + hi) * D + lanelo;
        acc0 = __builtin_amdgcn_wmma_f32_16x16x4_f32(false, a, false, bk[ 0], (short)0, acc0, false, false);
        acc1 = __builtin_amdgcn_wmma_f32_16x16x4_f32(false, a, false, bk[16], (short)0, acc1, false, false);
        acc2 = __builtin_amdgcn_wmma_f32_16x16x4_f32(false, a, false, bk[32], (short)0, acc2, false, false);
        acc3 = __builtin_amdgcn_wmma_f32_16x16x4_f32(false, a, false, bk[48], (short)0, acc3, false, false);
    }

    // Epilogue: bias (+ReLU), store. C/D layout: lane half -> rows M, M+8.
    const int rbase = tile * 16 + (hi << 3);
    const float bv0 = bias[ 0 + lanelo];
    const float bv1 = bias[16 + lanelo];
    const float bv2 = bias[32 + lanelo];
    const float bv3 = bias[48 + lanelo];
#pragma unroll
    for (int r = 0; r < 8; ++r) {
        float* orow = out + (size_t)(rbase + r) * D;
        float v0 = acc0[r] + bv0;
        float v1 = acc1[r] + bv1;
        float v2 = acc2[r] + bv2;
        float v3 = acc3[r] + bv3;
        if (relu) { v0 = fmaxf(v0, 0.f); v1 = fmaxf(v1, 0.f);
                    v2 = fmaxf(v2, 0.f); v3 = fmaxf(v3, 0.f); }
        orow[ 0 + lanelo] = v0;
        orow[16 + lanelo] = v1;
        orow[32 + lanelo] = v2;
        orow[48 + lanelo] = v3;
    }
}

// ---------------------------------------------------------------- launcher
extern "C" void kernel_launch(void* const* d_in, const int* in_sizes, int n_in,
                              void* d_out, int out_size, void* d_ws, size_t ws_size,
                              hipStream_t stream) {
    const float* x  = (const float*)d_in[0];
    const int*   ei = (const int*)  d_in[1];
    const float* W1 = (const float*)d_in[2];
    const float* b1 = (const float*)d_in[3];
    const float* W2 = (const float*)d_in[4];
    const float* b2 = (const float*)d_in[5];
    float* out = (float*)d_out;

    const int nNodes = in_sizes[0] / D;       // 100000
    const int nEdges = in_sizes[1] / 2;       // 1200000
    const int* row = ei;                      // edge_index[0]
    const int* col = ei + nEdges;             // edge_index[1]

    // Workspace layout: agg [N*64] | deg [N] | h [N*64]  (~51.6 MB)
    float* agg = (float*)d_ws;
    float* deg = agg + (size_t)nNodes * D;
    float* h   = deg + nNodes;

    const int nFeat = nNodes * D;                              // 6.4M
    const int scatterThreads = nEdges * 16;                    // 19.2M
    const int scatterBlocks  = (scatterThreads + 255) / 256;
    const int nTiles  = (nNodes + 15) / 16;                    // 6250 (exact)
    const int gBlocks = (nTiles + 7) / 8;                      // 8 waves/block

    // ---- layer 1 ----
    { int n = nFeat + nNodes;                                  // agg + deg
      zero_f32<<<(n + 255) / 256, 256, 0, stream>>>(agg, n); }
    sage_scatter<<<scatterBlocks, 256, 0, stream>>>(x, row, col, agg, deg, nEdges);
    sage_gemm_wmma<<<gBlocks, 256, 0, stream>>>(x, agg, deg, W1, b1, h, nTiles, /*relu=*/1);

    // ---- layer 2 (deg unchanged, reuse) ----
    zero_f32<<<(nFeat + 255) / 256, 256, 0, stream>>>(agg, nFeat);
    sage_scatter<<<scatterBlocks, 256, 0, stream>>>(h, row, col, agg, nullptr, nEdges);
    sage_gemm_wmma<<<gBlocks, 256, 0, stream>>>(h, agg, deg, W2, b2, out, nTiles, /*relu=*/0);
}